// SDConv_9560597201099
// MI455X (gfx1250) — compile-verified
//
#include <hip/hip_runtime.h>

typedef __attribute__((ext_vector_type(2))) float v2f;
typedef __attribute__((ext_vector_type(8))) float v8f;

// ---------------------------------------------------------------------------
// Phase 1: XW = X @ W_k for X in {X_real, X_imag} (blockIdx.y selects which),
// written interleaved as (re,im) pairs: out_pair[(n*64+col)*2 + comp].
// X: (N,64) f32 row-major, W: (64,64) f32 row-major.
// Uses V_WMMA_F32_16X16X4_F32; one wave computes a 16x16 tile, K-loop of 16.
// ---------------------------------------------------------------------------
__global__ __launch_bounds__(256) void gemm_xw_kernel(
    const float* __restrict__ Xr, const float* __restrict__ Xi,
    const float* __restrict__ W, float* __restrict__ out_pair, int N)
{
    __shared__ float lds[32 * 66];          // 32-row strip, stride 66 (pad -> no bank conflicts)
    const int comp = blockIdx.y;            // 0 = real X, 1 = imag X
    const float* __restrict__ X = comp ? Xi : Xr;
    const int row0 = blockIdx.x * 32;

    // Cooperative, fully coalesced load of the 32x64 strip into LDS (clamped).
    #pragma unroll
    for (int i = 0; i < 8; ++i) {
        int idx = i * 256 + threadIdx.x;
        int r = idx >> 6, c = idx & 63;
        int gr = row0 + r;
        if (gr >= N) gr = N - 1;
        lds[r * 66 + c] = X[(size_t)gr * 64 + c];
    }
    __syncthreads();

    const int wave    = threadIdx.x >> 5;
    const int lane    = threadIdx.x & 31;
    const int lane_lo = lane & 15;          // M (for A), N-col (for B/C/D)
    const int lane_hi = lane >> 4;          // selects K pair / upper-M half
    const int strip   = wave >> 2;          // 0..1: which 16-row strip
    const int ct      = wave & 3;           // 0..3: which 16-col tile

    v8f acc = {};
    const float* __restrict__ lrow = &lds[(strip * 16 + lane_lo) * 66];
    const int col = ct * 16 + lane_lo;

    #pragma unroll
    for (int kk = 0; kk < 16; ++kk) {
        const int kb = kk * 4 + lane_hi * 2;           // K base for this lane half
        v2f a; a.x = lrow[kb];                         // A[M=lane_lo][kb], [kb+1]
               a.y = lrow[kb + 1];
        v2f b; b.x = W[kb * 64 + col];                 // B[K=kb][N=col]
               b.y = W[(kb + 1) * 64 + col];
        acc = __builtin_amdgcn_wmma_f32_16x16x4_f32(
            /*neg_a=*/false, a, /*neg_b=*/false, b,
            /*c_mod=*/(short)0, acc, /*reuse_a=*/false, /*reuse_b=*/false);
    }

    // D layout: VGPR v -> M = v (lanes 0-15) / v+8 (lanes 16-31)
    #pragma unroll
    for (int v = 0; v < 8; ++v) {
        int row = row0 + strip * 16 + v + lane_hi * 8;
        if (row < N) out_pair[((size_t)row * 64 + col) * 2 + comp] = acc[v];
    }
}

// ---------------------------------------------------------------------------
// Phase 2: per-row segmented gather-accumulate (edge_row is sorted).
// One 64-thread group owns one output row; thread = channel. Segment bounds
// found by binary search -> no atomics, single deterministic store per row.
//   real[r] (+)= sum_e lr[e]*XWre[col[e]] - li[e]*XWim[col[e]]
//   imag[r] (+)= sum_e li[e]*XWre[col[e]] + lr[e]*XWim[col[e]]
// k==0 seeds with bias; k==1 accumulates onto previous output.
// ---------------------------------------------------------------------------
__global__ __launch_bounds__(256) void spmm_rows_kernel(
    const v2f* __restrict__ XW,            // (N*64) interleaved (re,im) pairs
    const float* __restrict__ lr, const float* __restrict__ li,
    const int* __restrict__ erow, const int* __restrict__ ecol,
    const float* __restrict__ bias,
    float* __restrict__ out_real, float* __restrict__ out_imag,
    int N, int E, int accumulate)
{
    const int row = blockIdx.x * 4 + (threadIdx.x >> 6);
    const int ch  = threadIdx.x & 63;
    if (row >= N) return;

    // lower_bound(erow, row) and lower_bound(erow, row+1)
    int l = 0, h = E;
    while (l < h) { int m = (l + h) >> 1; if (erow[m] < row) l = m + 1; else h = m; }
    const int lo = l;
    h = E;
    while (l < h) { int m = (l + h) >> 1; if (erow[m] < row + 1) l = m + 1; else h = m; }
    const int hi = l;

    const size_t o = (size_t)row * 64 + ch;
    float racc, iacc;
    if (accumulate) { racc = out_real[o]; iacc = out_imag[o]; }
    else            { racc = bias[ch];    iacc = bias[ch];    }

    for (int e = lo; e < hi; ++e) {
        const int c  = ecol[e];
        const float vr = lr[e], vi = li[e];
        const v2f ab = XW[(size_t)c * 64 + ch];        // coalesced 512B gather per edge
        racc += vr * ab.x - vi * ab.y;
        iacc += vi * ab.x + vr * ab.y;
    }
    out_real[o] = racc;
    out_imag[o] = iacc;
}

// ---------------------------------------------------------------------------
// Host launch: for k in {0,1}: GEMM (X@Wk -> ws pairs), then row scatter.
// Workspace use: N*64*2 floats = 51.2 MB (XW pairs), reused across k.
// ---------------------------------------------------------------------------
extern "C" void kernel_launch(void* const* d_in, const int* in_sizes, int n_in,
                              void* d_out, int out_size, void* d_ws, size_t ws_size,
                              hipStream_t stream) {
    const float* Xr   = (const float*)d_in[0];
    const float* Xi   = (const float*)d_in[1];
    const float* Lr   = (const float*)d_in[2];   // (2,E)
    const float* Li   = (const float*)d_in[3];   // (2,E)
    const float* W    = (const float*)d_in[4];   // (2,64,64)
    const float* bias = (const float*)d_in[5];   // (64)
    const int*   erow = (const int*)d_in[6];
    const int*   ecol = (const int*)d_in[7];

    const int E = in_sizes[6];
    const int N = in_sizes[0] / 64;

    float* out_real = (float*)d_out;
    float* out_imag = out_real + (size_t)N * 64;
    float* ws_pair  = (float*)d_ws;              // N*64*2 floats

    dim3 gemm_grid((N + 31) / 32, 2);
    const int spmm_blocks = (N + 3) / 4;

    for (int k = 0; k < 2; ++k) {
        gemm_xw_kernel<<<gemm_grid, 256, 0, stream>>>(
            Xr, Xi, W + (size_t)k * 64 * 64, ws_pair, N);
        spmm_rows_kernel<<<spmm_blocks, 256, 0, stream>>>(
            (const v2f*)ws_pair,
            Lr + (size_t)k * E, Li + (size_t)k * E,
            erow, ecol, bias, out_real, out_imag, N, E, /*accumulate=*/k);
    }
}